// NFPN_RPS_Net_42906723287494
// MI455X (gfx1250) — compile-verified
//
#include <hip/hip_runtime.h>

// NFPN fixed-point network, MI455X (gfx1250, wave32).
// One wave handles 16 rows; lanes r and r+16 both own row r's full state so
// WMMA A-operands are built with pure per-lane selects (no cross-lane ops).
// fc2/fc3 run on V_WMMA_F32_16X16X4_F32; C->A relayout goes through per-wave
// private LDS (in-order per wave => no barriers, only scheduling fences).

typedef float v2f __attribute__((ext_vector_type(2)));
typedef float v8f __attribute__((ext_vector_type(8)));

#define NFPN_EPS2   1e-10f   // (1e-5)^2
#define NFPN_MAXD   100
#define NFPN_SLOPE  0.1f

#define H_STRIDE 34          // 16x32 h tile, padded stride (bank-conflict free)
#define G_STRIDE 18          // 16x6 g tile, padded stride (bank-conflict free)
#define WAVE_LDS (16*H_STRIDE + 16*G_STRIDE)

__device__ __forceinline__ void comp_swap(float& a, float& b) {
  float mx = fmaxf(a, b);
  float mn = fminf(a, b);
  a = mx; b = mn;                       // descending order
}

// One latent-space step: un = Proj_simplex( -(fc3(leaky(fc2([u,Qd]))) + b3) )
__device__ __forceinline__ void latent_step(
    const float* __restrict__ u, const float* __restrict__ qd,
    const v2f* __restrict__ w2b,       // [2 n-tiles][3 k-steps] flattened
    const v2f* __restrict__ w3b,       // [8 k-steps]
    float b2s0, float b2s1, float b3s,
    float* hbase, float* gbase,
    int r, int hi, float* __restrict__ un)
{
  // A operands for fc2: x = [u(6), Qd(6)] as 16x12, 3 k-steps of 4.
  // A layout (16x4 f32): lanes 0-15 hold K=4s,4s+1; lanes 16-31 hold K=4s+2,4s+3.
  v2f a0, a1, a2;
  a0.x = hi ? u[2]  : u[0];   a0.y = hi ? u[3]  : u[1];
  a1.x = hi ? qd[0] : u[4];   a1.y = hi ? qd[1] : u[5];
  a2.x = hi ? qd[4] : qd[2];  a2.y = hi ? qd[5] : qd[3];

  v8f acc0 = {0.f,0.f,0.f,0.f,0.f,0.f,0.f,0.f};
  v8f acc1 = {0.f,0.f,0.f,0.f,0.f,0.f,0.f,0.f};
  acc0 = __builtin_amdgcn_wmma_f32_16x16x4_f32(false, a0, false, w2b[0], (short)0, acc0, false, false);
  acc0 = __builtin_amdgcn_wmma_f32_16x16x4_f32(false, a1, false, w2b[1], (short)0, acc0, false, false);
  acc0 = __builtin_amdgcn_wmma_f32_16x16x4_f32(false, a2, false, w2b[2], (short)0, acc0, false, false);
  acc1 = __builtin_amdgcn_wmma_f32_16x16x4_f32(false, a0, false, w2b[3], (short)0, acc1, false, false);
  acc1 = __builtin_amdgcn_wmma_f32_16x16x4_f32(false, a1, false, w2b[4], (short)0, acc1, false, false);
  acc1 = __builtin_amdgcn_wmma_f32_16x16x4_f32(false, a2, false, w2b[5], (short)0, acc1, false, false);

  // bias + LeakyReLU in C-layout, scatter h[16][32] row-major into LDS.
  // C layout: VGPR j, lane = n + 16*(m>=8): element (m = j+8*hi, n = r [+16t]).
#pragma unroll
  for (int j = 0; j < 8; ++j) {
    int m = j + 8*hi;
    float h0 = acc0[j] + b2s0;
    h0 = h0 > 0.f ? h0 : NFPN_SLOPE * h0;
    float h1 = acc1[j] + b2s1;
    h1 = h1 > 0.f ? h1 : NFPN_SLOPE * h1;     // cols >=30 are exactly 0
    hbase[m*H_STRIDE + r]      = h0;
    hbase[m*H_STRIDE + 16 + r] = h1;
  }
  __builtin_amdgcn_wave_barrier();            // LDS is in-order per wave

  // fc3: gather h rows back in A-layout (8 k-steps over K=32, zero-padded).
  v8f g = {0.f,0.f,0.f,0.f,0.f,0.f,0.f,0.f};
#pragma unroll
  for (int s = 0; s < 8; ++s) {
    const v2f a = *(const v2f*)(hbase + r*H_STRIDE + 4*s + 2*hi);  // 8B aligned
    g = __builtin_amdgcn_wmma_f32_16x16x4_f32(false, a, false, w3b[s], (short)0, g, false, false);
  }

  // t = u - Fxd = -(g + b3); scatter then per-lane gather of this row's 6 vals.
#pragma unroll
  for (int j = 0; j < 8; ++j) {
    int m = j + 8*hi;
    gbase[m*G_STRIDE + r] = -(g[j] + b3s);
  }
  __builtin_amdgcn_wave_barrier();

  float v[6];
#pragma unroll
  for (int c = 0; c < 6; ++c) v[c] = gbase[r*G_STRIDE + c];

  // Euclidean projection onto the simplex: sort desc (12-comparator network).
  float s0=v[0], s1=v[1], s2=v[2], s3=v[3], s4=v[4], s5=v[5];
  comp_swap(s0,s5); comp_swap(s1,s3); comp_swap(s2,s4);
  comp_swap(s1,s2); comp_swap(s3,s4);
  comp_swap(s0,s3); comp_swap(s2,s5);
  comp_swap(s0,s1); comp_swap(s2,s3); comp_swap(s4,s5);
  comp_swap(s1,s2); comp_swap(s3,s4);
  float srt[6] = {s0,s1,s2,s3,s4,s5};

  // theta = (cumsum-1)/rho with rho = last k+1 where s[k]*(k+1) > cumsum-1.
  // Divisors are the constants 1..6 -> multiply by reciprocal (<=1ulp vs div).
  const float rcpk[6] = {1.f, 0.5f, 1.f/3.f, 0.25f, 0.2f, 1.f/6.f};
  float css = -1.f, theta = 0.f;
#pragma unroll
  for (int k = 0; k < 6; ++k) {
    css += srt[k];
    float kk = (float)(k + 1);
    if (srt[k]*kk > css) theta = css * rcpk[k];  // prefix-true => last true wins
  }
#pragma unroll
  for (int c = 0; c < 6; ++c) un[c] = fmaxf(v[c] - theta, 0.f);
}

__global__ __launch_bounds__(256) void nfpn_fixed_point_kernel(
    const float* __restrict__ D,
    const float* __restrict__ W1, const float* __restrict__ b1,
    const float* __restrict__ W2, const float* __restrict__ b2,
    const float* __restrict__ W3, const float* __restrict__ b3,
    float* __restrict__ out, int nrows)
{
  __shared__ float smem[8 * WAVE_LDS];        // 26.6 KB / block of 8 waves
  const int tid  = threadIdx.x;
  const int wave = tid >> 5;
  const int lane = tid & 31;
  const int r    = lane & 15;
  const int hi   = lane >> 4;

  const long waveBase = (long)blockIdx.x * 128 + (long)wave * 16;
  if (waveBase >= nrows) return;              // wave-uniform exit
  const long row  = waveBase + r;
  const long rowc = row < nrows ? row : (long)(nrows - 1);

  float* hbase = smem + wave * WAVE_LDS;
  float* gbase = hbase + 16 * H_STRIDE;

  // Qd = d @ W1^T + b1 (per-lane, 18 FMAs, done once)
  const float d0 = D[rowc*3+0], d1 = D[rowc*3+1], d2 = D[rowc*3+2];
  float qd[6];
#pragma unroll
  for (int i = 0; i < 6; ++i)
    qd[i] = b1[i] + W1[i*3+0]*d0 + W1[i*3+1]*d1 + W1[i*3+2]*d2;

  // B operands, resident in registers for the whole iteration.
  // B layout (4x16 f32): VGPR0/1 = K=(2*hi)+0/1 rows, N = lane&15.
  v2f w2b[6];
#pragma unroll
  for (int t = 0; t < 2; ++t) {
    const int n = 16*t + r;
#pragma unroll
    for (int s = 0; s < 3; ++s) {
      const int k = 4*s + 2*hi;
      w2b[t*3+s].x = (n < 30) ? W2[n*12 + k]     : 0.f;
      w2b[t*3+s].y = (n < 30) ? W2[n*12 + k + 1] : 0.f;
    }
  }
  v2f w3b[8];
#pragma unroll
  for (int s = 0; s < 8; ++s) {
    const int k = 4*s + 2*hi;
    w3b[s].x = (r < 6 && k     < 30) ? W3[r*30 + k]     : 0.f;
    w3b[s].y = (r < 6 && k + 1 < 30) ? W3[r*30 + k + 1] : 0.f;
  }
  const float b2s0 = b2[r];
  const float b2s1 = (16 + r < 30) ? b2[16 + r] : 0.f;
  const float b3s  = (r < 6) ? b3[r] : 0.f;

  // Fixed-point loop: u0 = 0, u_prev = inf => at least one iteration.
  // Wave-level convergence (ballot) keeps EXEC all-ones for every WMMA.
  float u[6] = {0.f, 0.f, 0.f, 0.f, 0.f, 0.f};
  int depth = 0;
  while (true) {
    float un[6];
    latent_step(u, qd, w2b, w3b, b2s0, b2s1, b3s, hbase, gbase, r, hi, un);
    float res2 = 0.f;
#pragma unroll
    for (int c = 0; c < 6; ++c) {
      const float dd = un[c] - u[c];
      res2 += dd * dd;
      u[c] = un[c];
    }
    ++depth;
    const unsigned live = __builtin_amdgcn_ballot_w32(res2 > NFPN_EPS2);
    if (live == 0u || depth >= NFPN_MAXD) break;
  }

  // One gradient-attached step on the (detached) fixed point = the output.
  float uf[6];
  latent_step(u, qd, w2b, w3b, b2s0, b2s1, b3s, hbase, gbase, r, hi, uf);

  if (hi == 0 && row < nrows) {
#pragma unroll
    for (int c = 0; c < 6; ++c) out[row*6 + c] = uf[c];
  }
}

extern "C" void kernel_launch(void* const* d_in, const int* in_sizes, int n_in,
                              void* d_out, int out_size, void* d_ws, size_t ws_size,
                              hipStream_t stream) {
  const float* D  = (const float*)d_in[0];
  const float* W1 = (const float*)d_in[1];
  const float* b1 = (const float*)d_in[2];
  const float* W2 = (const float*)d_in[3];
  const float* b2 = (const float*)d_in[4];
  const float* W3 = (const float*)d_in[5];
  const float* b3 = (const float*)d_in[6];
  float* out = (float*)d_out;

  const int nrows = in_sizes[0] / 3;          // d is [B, 3]
  const int grid  = (nrows + 127) / 128;      // 128 rows per 256-thread block
  nfpn_fixed_point_kernel<<<grid, 256, 0, stream>>>(D, W1, b1, W2, b2, W3, b3,
                                                    out, nrows);
  (void)n_in; (void)out_size; (void)d_ws; (void)ws_size;
}